// DeltaAI_34703335752317
// MI455X (gfx1250) — compile-verified
//
#include <hip/hip_runtime.h>
#include <hip/hip_bf16.h>

#define VDIM 2048
#define HDIM 64
#define LN_EPS 1e-5f
#define ROWS_PER_WG 64
#define KC 64
#define NCHUNK (VDIM / KC)
#define PITCH 68      // rows 16B-aligned (68*4=272 = 17*16), good LDS bank spread
#define NTHREADS 128

#if defined(__AMDGCN__) && __has_builtin(__builtin_amdgcn_global_load_async_to_lds_b128)
#define HAVE_ASYNC_LDS 1
#else
#define HAVE_ASYNC_LDS 0
#endif

typedef float v2f __attribute__((ext_vector_type(2)));
typedef float v8f __attribute__((ext_vector_type(8)));
#if HAVE_ASYNC_LDS
typedef int v4i __attribute__((vector_size(4 * sizeof(int))));
typedef __attribute__((address_space(1))) v4i* gv4i_p;  // global int32x4*
typedef __attribute__((address_space(3))) v4i* lv4i_p;  // LDS int32x4*
#endif

__device__ __forceinline__ float allreduce16_add(float v) {
  // butterfly over masks 1,2,4,8 stays inside each 16-lane half of the wave32,
  // which is exactly the column span of one row in the WMMA C/D layout.
  v += __shfl_xor(v, 1, 32);
  v += __shfl_xor(v, 2, 32);
  v += __shfl_xor(v, 4, 32);
  v += __shfl_xor(v, 8, 32);
  return v;
}

__device__ __forceinline__ void wait_async_done() {
#if HAVE_ASYNC_LDS
#if __has_builtin(__builtin_amdgcn_s_wait_asynccnt)
  __builtin_amdgcn_s_wait_asynccnt(0);
#else
  asm volatile("s_wait_asynccnt 0x0" ::: "memory");
#endif
#endif
}

// Stage one 64x64 V tile and one 64x64 W1 chunk (both row-major, pitch 68)
// into LDS. Async direct-to-LDS b128 when available; sync float4 copy otherwise.
__device__ __forceinline__ void stage_chunk(const float* __restrict__ Vblk,
                                            const float* __restrict__ Wg, int kc,
                                            float* __restrict__ vbuf,
                                            float* __restrict__ wbuf, int tid) {
  const int vrow = tid & 63;
  const int vcol0 = (tid >> 6) * 32;
  const float* vsrc = Vblk + (size_t)vrow * VDIM + kc + vcol0;
  float* vdst = vbuf + vrow * PITCH + vcol0;
#if HAVE_ASYNC_LDS
#pragma unroll
  for (int i = 0; i < 32; i += 4)
    __builtin_amdgcn_global_load_async_to_lds_b128(
        (gv4i_p)(vsrc + i), (lv4i_p)(vdst + i), 0, 0);
#pragma unroll
  for (int i = 0; i < 8; ++i) {
    int f = i * NTHREADS + tid;  // float4 index within the 64x64 chunk
    int k = f >> 4;
    int n = (f & 15) * 4;
    __builtin_amdgcn_global_load_async_to_lds_b128(
        (gv4i_p)(Wg + (size_t)(kc + k) * HDIM + n),
        (lv4i_p)(wbuf + k * PITCH + n), 0, 0);
  }
#else
#pragma unroll
  for (int i = 0; i < 32; i += 4) {
    float4 v4 = *(const float4*)(vsrc + i);
    *(float4*)(vdst + i) = v4;
  }
#pragma unroll
  for (int i = 0; i < 8; ++i) {
    int f = i * NTHREADS + tid;
    int k = f >> 4;
    int n = (f & 15) * 4;
    float4 w4 = *(const float4*)(Wg + (size_t)(kc + k) * HDIM + n);
    *(float4*)(wbuf + k * PITCH + n) = w4;
  }
#endif
}

// One wave accumulates a 16x64 f32 tile with 4 accumulators of
// v_wmma_f32_16x16x4_f32.
//   As = &vbuf[(wave*16+lo)*PITCH + 2*hi]   (A layout: lane lo = row)
//   Bs = &wbuf[2*hi*PITCH + lo]             (B from natural [k][n] layout)
// Pointers MUST be derived directly from __shared__ arrays at each call site
// so address-space inference keeps these as DS ops (no private pointer
// arrays -> no flat loads).
// ACC_ABS folds |A| accumulation into the fragment loads (each lane sees
// exactly half its row's K elements; halves are complementary).
template <bool ACC_ABS>
__device__ __forceinline__ void wmma_tile64(v8f acc[4], const float* __restrict__ As,
                                            const float* __restrict__ Bs, int kLen,
                                            float* __restrict__ absAcc) {
#pragma unroll 4
  for (int k = 0; k < kLen; k += 4) {
    v2f a;
    a.x = As[k];
    a.y = As[k + 1];
    if (ACC_ABS) *absAcc += fabsf(a.x) + fabsf(a.y);
#pragma unroll
    for (int t = 0; t < 4; ++t) {
      v2f b;
      b.x = Bs[k * PITCH + t * 16];
      b.y = Bs[(k + 1) * PITCH + t * 16];
      acc[t] = __builtin_amdgcn_wmma_f32_16x16x4_f32(
          false, a, false, b, (short)0, acc[t], false, false);
    }
  }
}

// bias add -> LayerNorm over 64 columns -> ReLU, in WMMA C/D register layout.
__device__ __forceinline__ void bias_ln_relu(v8f acc[4], const float* __restrict__ bias,
                                             const float* __restrict__ g,
                                             const float* __restrict__ be, int lo) {
  float bb[4], gg[4], ee[4];
#pragma unroll
  for (int t = 0; t < 4; ++t) {
    bb[t] = bias[t * 16 + lo];
    gg[t] = g[t * 16 + lo];
    ee[t] = be[t * 16 + lo];
  }
#pragma unroll
  for (int t = 0; t < 4; ++t)
#pragma unroll
    for (int j = 0; j < 8; ++j) acc[t][j] += bb[t];
#pragma unroll
  for (int j = 0; j < 8; ++j) {
    float s = acc[0][j] + acc[1][j] + acc[2][j] + acc[3][j];
    float q = acc[0][j] * acc[0][j] + acc[1][j] * acc[1][j] +
              acc[2][j] * acc[2][j] + acc[3][j] * acc[3][j];
    s = allreduce16_add(s);
    q = allreduce16_add(q);
    float mu  = s * (1.0f / 64.0f);
    float var = q * (1.0f / 64.0f) - mu * mu;
    float inv = rsqrtf(var + LN_EPS);
#pragma unroll
    for (int t = 0; t < 4; ++t) {
      float x = (acc[t][j] - mu) * inv * gg[t] + ee[t];
      acc[t][j] = fmaxf(x, 0.0f);
    }
  }
}

__global__ __launch_bounds__(NTHREADS) void deltaai_fused(
    const float* __restrict__ V, const int* __restrict__ ilist,
    const int* __restrict__ temp,
    const float* __restrict__ W1, const float* __restrict__ b1,
    const float* __restrict__ g1, const float* __restrict__ be1,
    const float* __restrict__ W2, const float* __restrict__ b2,
    const float* __restrict__ g2, const float* __restrict__ be2,
    const float* __restrict__ W3, const float* __restrict__ b3,
    const float* __restrict__ g3, const float* __restrict__ be3,
    const float* __restrict__ Wp, const float* __restrict__ bp,
    const float* __restrict__ marginals, float* __restrict__ out) {
  __shared__ float vbufs[2][ROWS_PER_WG * PITCH];  // double-buffered V tiles
  __shared__ float wbufs[2][HDIM * PITCH];         // double-buffered W1 chunks

  const int tid  = threadIdx.x;
  const int lane = tid & 31;
  const int wave = tid >> 5;
  const int lo   = lane & 15;
  const int hi   = lane >> 4;
  const int row0 = blockIdx.x * ROWS_PER_WG;

  v8f acc[4];
#pragma unroll
  for (int t = 0; t < 4; ++t)
#pragma unroll
    for (int j = 0; j < 8; ++j) acc[t][j] = 0.0f;

  const float* Vblk = V + (size_t)row0 * VDIM;
  const int aOff = (wave * 16 + lo) * PITCH + 2 * hi;
  const int bOff = 2 * hi * PITCH + lo;
  float absLocal = 0.0f;

  // ---- layer 1: [16x2048]@[2048x64] per wave; double-buffered staging.
  // Pipeline unrolled by 2 so each half names its LDS buffer directly.
  stage_chunk(Vblk, W1, 0, vbufs[0], wbufs[0], tid);
#pragma unroll 1
  for (int ic = 0; ic < NCHUNK; ic += 2) {
    wait_async_done();   // my async loads for chunk ic have landed in LDS
    __syncthreads();     // everyone's chunk ic staged; prev compute done
    if (ic + 1 < NCHUNK)
      stage_chunk(Vblk, W1, (ic + 1) * KC, vbufs[1], wbufs[1], tid);
    wmma_tile64<true>(acc, &vbufs[0][aOff], &wbufs[0][bOff], KC, &absLocal);

    wait_async_done();   // chunk ic+1 staged
    __syncthreads();
    if (ic + 2 < NCHUNK)
      stage_chunk(Vblk, W1, (ic + 2) * KC, vbufs[0], wbufs[0], tid);
    wmma_tile64<true>(acc, &vbufs[1][aOff], &wbufs[1][bOff], KC, &absLocal);
  }
  // total |V| per row: lanes l and l+16 hold complementary halves of row lo
  const float rowAbsFull = absLocal + __shfl_xor(absLocal, 16, 32);

  __syncthreads();  // all waves done with both buffers

  // stage W2 (row-major copy) into wbufs[0] while epilogue runs
  for (int e = tid; e < HDIM * HDIM; e += NTHREADS) {
    int k = e >> 6, n = e & 63;
    wbufs[0][k * PITCH + n] = W2[e];
  }

  bias_ln_relu(acc, b1, g1, be1, lo);
  float* xw = &vbufs[0][wave * 16 * PITCH];  // per-wave tile: C/D -> A layout
#pragma unroll
  for (int t = 0; t < 4; ++t)
#pragma unroll
    for (int j = 0; j < 8; ++j)
      xw[(j + 8 * hi) * PITCH + t * 16 + lo] = acc[t][j];
  __syncthreads();  // W2 visible

  // ---- layer 2 ----
  v8f acc2[4];
#pragma unroll
  for (int t = 0; t < 4; ++t)
#pragma unroll
    for (int j = 0; j < 8; ++j) acc2[t][j] = 0.0f;
  wmma_tile64<false>(acc2, &vbufs[0][aOff], &wbufs[0][bOff], HDIM, nullptr);
  __syncthreads();  // W2 reads done

  for (int e = tid; e < HDIM * HDIM; e += NTHREADS) {
    int k = e >> 6, n = e & 63;
    wbufs[0][k * PITCH + n] = W3[e];
  }
  bias_ln_relu(acc2, b2, g2, be2, lo);
#pragma unroll
  for (int t = 0; t < 4; ++t)
#pragma unroll
    for (int j = 0; j < 8; ++j)
      xw[(j + 8 * hi) * PITCH + t * 16 + lo] = acc2[t][j];
  __syncthreads();  // W3 visible

  // ---- layer 3 ----
#pragma unroll
  for (int t = 0; t < 4; ++t)
#pragma unroll
    for (int j = 0; j < 8; ++j) acc[t][j] = 0.0f;
  wmma_tile64<false>(acc, &vbufs[0][aOff], &wbufs[0][bOff], HDIM, nullptr);
  bias_ln_relu(acc, b3, g3, be3, lo);
#pragma unroll
  for (int t = 0; t < 4; ++t)
#pragma unroll
    for (int j = 0; j < 8; ++j)
      xw[(j + 8 * hi) * PITCH + t * 16 + lo] = acc[t][j];
  // per-wave LDS ops are in-order: safe to read xw below without a barrier

  // ---- gathered per-variable head ----
  const int rowG = row0 + wave * 16 + lo;
  const int idx  = ilist[rowG];
  const float* wp = Wp + (size_t)idx * HDIM + hi * 32;
  const float* xr = &vbufs[0][(wave * 16 + lo) * PITCH + hi * 32];
  float dot = 0.0f;
#pragma unroll
  for (int c = 0; c < 32; ++c) dot = fmaf(xr[c], wp[c], dot);
  dot += __shfl_xor(dot, 16, 32);  // combine the two half-row partials
  if (hi == 0) {
    float res = dot + bp[idx];
    if (rowAbsFull == 0.0f) res = marginals[idx];
    out[rowG] = res / (float)temp[0];
  }
}

extern "C" void kernel_launch(void* const* d_in, const int* in_sizes, int n_in,
                              void* d_out, int out_size, void* d_ws, size_t ws_size,
                              hipStream_t stream) {
  (void)n_in; (void)d_ws; (void)ws_size; (void)out_size;
  const float* V    = (const float*)d_in[0];
  const int*   il   = (const int*)d_in[1];
  const int*   temp = (const int*)d_in[2];
  const float* W1   = (const float*)d_in[3];
  const float* b1   = (const float*)d_in[4];
  const float* g1   = (const float*)d_in[5];
  const float* be1  = (const float*)d_in[6];
  const float* W2   = (const float*)d_in[7];
  const float* b2   = (const float*)d_in[8];
  const float* g2   = (const float*)d_in[9];
  const float* be2  = (const float*)d_in[10];
  const float* W3   = (const float*)d_in[11];
  const float* b3   = (const float*)d_in[12];
  const float* g3   = (const float*)d_in[13];
  const float* be3  = (const float*)d_in[14];
  const float* Wp   = (const float*)d_in[15];
  const float* bp   = (const float*)d_in[16];
  const float* mg   = (const float*)d_in[17];
  float* out = (float*)d_out;

  const int n = in_sizes[1];         // N rows
  const int grid = n / ROWS_PER_WG;  // 65536/64 = 1024
  deltaai_fused<<<dim3(grid), dim3(NTHREADS), 0, stream>>>(
      V, il, temp, W1, b1, g1, be1, W2, b2, g2, be2,
      W3, b3, g3, be3, Wp, bp, mg, out);
}